// roughBergomi_69277822484667
// MI455X (gfx1250) — compile-verified
//
#include <hip/hip_runtime.h>
#include <hip/hip_bf16.h>

typedef __attribute__((ext_vector_type(2))) float v2f;
typedef __attribute__((ext_vector_type(8))) float v8f;

// Problem shape is fixed by the harness: T = 100 steps, output T+1 = 101.
constexpr int T   = 100;
constexpr int TP1 = 101;
constexpr int SXS = 116;              // LDS row stride (floats): 116 % 64 = 52 -> conflict-light
constexpr int PATHS_PER_BLOCK = 128;  // 8 wave32 x 16-path WMMA tiles
constexpr int THREADS = 256;
constexpr int NTILES = 7;             // ceil(112/16) output tiles (cols 101..111 are discarded pad)

__global__ __launch_bounds__(THREADS)
void roughBergomi_wmma_kernel(const float* __restrict__ z1,
                              const float* __restrict__ dw2,
                              const float* __restrict__ pr,
                              const float* __restrict__ pa,
                              const float* __restrict__ prho,
                              const float* __restrict__ pxi,
                              const float* __restrict__ peta,
                              const float* __restrict__ ps0,
                              const int*   __restrict__ psteps,
                              float* __restrict__ S,
                              int N)
{
    __shared__ float sX[PATHS_PER_BLOCK * SXS]; // dw1_0 rows, later overwritten with conv rows
    __shared__ float sGrev[224];                // REVERSED G: sGrev[i] = G[123 - i]; zeros outside
    __shared__ float sTpow[112];                // (j*dt)^(2a+1)

    const int tid = threadIdx.x;
    const int blockBase = blockIdx.x * PATHS_PER_BLOCK;

    // ---- scalar parameters (1-element device arrays) ----
    const float r   = pr[0];
    const float a   = pa[0];
    const float rho = prho[0];
    const float xi  = pxi[0];
    const float eta = peta[0];
    const float s0  = ps0[0];
    const float steps_f = (float)psteps[0];

    const float dt   = 1.0f / steps_f;
    const float ap1  = a + 1.0f;
    const float a2p1 = 2.0f * a + 1.0f;
    const float c00  = 1.0f / steps_f;
    const float c01  = 1.0f / (ap1 * powf(steps_f, ap1));
    const float c11  = 1.0f / (a2p1 * powf(steps_f, a2p1));
    const float l00  = sqrtf(c00);
    const float l10  = c01 / l00;
    const float l11  = sqrtf(c11 - l10 * l10);
    const float invStepsA = 1.0f / powf(steps_f, a);   // steps^{-a}

    // ---- Phase 1: stage dw1_0 = z1[...,0]*l00 into LDS; build reversed-G pad + tpow tables ----
    for (int i = tid; i < PATHS_PER_BLOCK * T; i += THREADS) {
        int p = i / T;
        int t = i - p * T;
        int gp = blockBase + p;
        if (gp >= N) gp = N - 1;                       // clamp for tail block (data unused)
        sX[p * SXS + t] = z1[(size_t)(gp * T + t) * 2] * l00;
    }
    for (int i = tid; i < 224; i += THREADS) {
        float v = 0.0f;
        int j = 123 - i;                               // G index; G[0]=G[1]=0
        if (j >= 2 && j <= T) {
            float fj = (float)j;
            // (bk/steps)^a simplifies: bk = X^{1/a} => (bk/steps)^a = X * steps^{-a}
            v = (powf(fj, ap1) - powf(fj - 1.0f, ap1)) / ap1 * invStepsA;
        }
        sGrev[i] = v;
    }
    for (int i = tid; i < 112; i += THREADS) {
        sTpow[i] = (i >= 1 && i <= T) ? powf((float)i * dt, a2p1) : 0.0f;
    }
    __syncthreads();

    // ---- Phase 2: conv = dw1_0 (16xK) x ToeplitzG (KxN) via V_WMMA_F32_16X16X4_F32 ----
    {
        const int lane = tid & 31;
        const int nLo  = lane & 15;    // A: row m ; B/D: col n
        const int half = lane >> 4;    // K (A/B) and M (D) lane-half split
        const int tileBase = (tid >> 5) * 16;   // this wave's 16 LDS rows

        v8f accs[NTILES];
        #pragma unroll
        for (int nt = 0; nt < NTILES; ++nt) {
            v8f c = {0.f, 0.f, 0.f, 0.f, 0.f, 0.f, 0.f, 0.f};
            const int nGlob = nt * 16 + nLo;
            for (int k0 = 0; k0 < T; k0 += 4) {
                // A fragment: A[m,k], vgpr = k&1, laneHalf = k>>1 (ascending pair load)
                v2f av;
                const float* rowA = &sX[(tileBase + nLo) * SXS + k0 + 2 * half];
                av.x = rowA[0];
                av.y = rowA[1];
                // B fragment: B[k,n] = G[nGlob - k] via reversed zero-padded table so the
                // two taps sit at ASCENDING consecutive LDS addresses -> no swap movs.
                const int q = 123 - nGlob + k0 + 2 * half; // sGrev[q] = G[nGlob-k0-2*half]
                v2f bv;
                bv.x = sGrev[q];
                bv.y = sGrev[q + 1];
                c = __builtin_amdgcn_wmma_f32_16x16x4_f32(
                        /*neg_a=*/false, av, /*neg_b=*/false, bv,
                        /*c_mod=*/(short)0, c, /*reuse_a=*/false, /*reuse_b=*/false);
            }
            accs[nt] = c;
        }
        // Write conv in-place over this wave's own rows (all reads done; no cross-wave rows)
        #pragma unroll
        for (int nt = 0; nt < NTILES; ++nt) {
            #pragma unroll
            for (int rI = 0; rI < 8; ++rI) {
                // D layout: m = rI + 8*half, n = nLo
                sX[(tileBase + rI + 8 * half) * SXS + nt * 16 + nLo] = accs[nt][rI];
            }
        }
    }
    __syncthreads();

    // ---- Phase 3: per-path variance process + log-price cumsum ----
    if (tid < PATHS_PER_BLOCK) {
        const int p  = blockBase + tid;
        const int pl = (p < N) ? p : N - 1;
        const bool wr = (p < N);
        const float sqdt  = sqrtf(dt);
        const float rhoc  = sqrtf(1.0f - rho * rho) * sqdt;
        const float sq2a1 = sqrtf(a2p1);
        const float he2   = 0.5f * eta * eta;

        float acc = 0.0f;        // cumsum of incr
        float prevDw11 = 0.0f;   // dw1_1[j-1]
        for (int j = 0; j <= T; ++j) {
            float Y1    = (j == 0) ? 0.0f : prevDw11;
            float convj = sX[tid * SXS + j];
            float Y     = sq2a1 * (Y1 + convj);
            float V     = xi * expf(eta * Y - he2 * sTpow[j]);
            if (wr) S[(size_t)p * TP1 + j] = s0 * expf(acc);
            if (j < T) {
                const float* zp = &z1[(size_t)(pl * T + j) * 2];
                float zx = zp[0];
                float zy = zp[1];
                float w2 = dw2[(size_t)pl * T + j];
                float dB = rho * (zx * l00) + rhoc * w2;
                acc += sqrtf(V) * dB + (r - 0.5f * V) * dt;
                prevDw11 = zx * l10 + zy * l11;
            }
        }
    }
}

extern "C" void kernel_launch(void* const* d_in, const int* in_sizes, int n_in,
                              void* d_out, int out_size, void* d_ws, size_t ws_size,
                              hipStream_t stream) {
    const float* z1    = (const float*)d_in[0];
    const float* dw2   = (const float*)d_in[1];
    const float* pr    = (const float*)d_in[2];
    const float* pa    = (const float*)d_in[3];
    const float* prho  = (const float*)d_in[4];
    const float* pxi   = (const float*)d_in[5];
    const float* peta  = (const float*)d_in[6];
    const float* ps0   = (const float*)d_in[7];
    const int*   psteps= (const int*)d_in[8];
    float* S = (float*)d_out;

    const int N = in_sizes[1] / T;   // dw2 is [N, T]
    const int blocks = (N + PATHS_PER_BLOCK - 1) / PATHS_PER_BLOCK;
    roughBergomi_wmma_kernel<<<blocks, THREADS, 0, stream>>>(
        z1, dw2, pr, pa, prho, pxi, peta, ps0, psteps, S, N);
}